// SoftThresholdingOperation_76879914598913
// MI455X (gfx1250) — compile-verified
//
#include <hip/hip_runtime.h>

#define ROWLEN 4096
#define NV4    (ROWLEN / 4)       // 1024 float4 per row
#define THREADS 256
#define PER_T  (ROWLEN / THREADS) // 16 scalars / thread
#define PER_T4 (NV4 / THREADS)    // 4 float4 / thread
#define KTOP   128
#define EPSF   1e-8f

typedef __attribute__((ext_vector_type(2))) float    v2f;
typedef __attribute__((ext_vector_type(4))) float    v4f;
typedef __attribute__((ext_vector_type(8))) float    v8f;
typedef __attribute__((ext_vector_type(4))) unsigned u32x4;
typedef __attribute__((ext_vector_type(4))) int      i32x4;
typedef __attribute__((ext_vector_type(8))) int      i32x8;
typedef int v4i_vs __attribute__((vector_size(4 * sizeof(int))));

#if defined(__HIP_DEVICE_COMPILE__) && __has_builtin(__builtin_amdgcn_wmma_f32_16x16x4_f32)
#define USE_WMMA_SCAN 1
#else
#define USE_WMMA_SCAN 0
#endif

#if defined(__HIP_DEVICE_COMPILE__) && __has_builtin(__builtin_amdgcn_global_load_async_to_lds_b128)
#define USE_ASYNC_LDS 1
#else
#define USE_ASYNC_LDS 0
#endif

// TDM builtin exists on both toolchains but with different arity:
//   ROCm 7.2 (clang-22):      5 args (u32x4, i32x8, i32x4, i32x4, i32)
//   amdgpu-toolchain (c-23):  6 args (u32x4, i32x8, i32x4, i32x4, i32x8, i32)
// The 6-arg toolchain ships amd_gfx1250_TDM.h -- use the header to pick arity.
#if defined(__HIP_DEVICE_COMPILE__) && USE_ASYNC_LDS && \
    __has_builtin(__builtin_amdgcn_tensor_load_to_lds) && \
    __has_builtin(__builtin_amdgcn_s_wait_tensorcnt)
#define USE_TDM 1
#if __has_include(<hip/amd_detail/amd_gfx1250_TDM.h>)
#define TDM_6ARG 1
#else
#define TDM_6ARG 0
#endif
#else
#define USE_TDM 0
#endif

#define AS1 __attribute__((address_space(1)))
#define AS3 __attribute__((address_space(3)))

// Monotonic float -> uint key: descending float order == descending unsigned order.
__device__ __forceinline__ unsigned f2key(float f) {
    unsigned u = __float_as_uint(f);
    return (u & 0x80000000u) ? ~u : (u | 0x80000000u);
}

__global__ __launch_bounds__(THREADS)
void soft_threshold_topk_kernel(const float* __restrict__ x, float* __restrict__ out) {
    __shared__ v4f      sv4[NV4];       // shifted row values (16B-aligned LDS)
    __shared__ unsigned hist[256];
    __shared__ unsigned cum[256];       // suffix-inclusive histogram scan
    __shared__ float    topv[KTOP];     // top-128, sorted descending
    __shared__ float    bpad[256];      // topv zero-padded to 16 columns (B fragment)
    __shared__ float    cs[KTOP];       // inclusive cumsum of topv
    __shared__ float    soff[8];        // segment offsets for WMMA scan
    __shared__ float    redbuf[8];
    __shared__ float    s_mx, s_tau;
    __shared__ unsigned s_selb, s_want, s_gtc, s_tiec, s_scount;

    const int tid = threadIdx.x;
    const long long row = blockIdx.x;
    const v4f* __restrict__ x4 = (const v4f*)(x + row * (long long)ROWLEN);
    v4f* __restrict__ o4       = (v4f*)(out + row * (long long)ROWLEN);
    float* sv = (float*)sv4;

    // ---- 1) Row -> LDS ---------------------------------------------------------
    v4f loc[PER_T4];
    float tmax = -3.402823466e38f;
#if USE_TDM
    // First 8KB: one Tensor Data Mover descriptor (TENSORcnt), issued by wave 0.
    // Second 8KB: per-lane async B128 DMA (ASYNCcnt). Both engines run concurrently.
    if (tid < 32) {
        const unsigned lds_base = (unsigned)(unsigned long long)(AS3 char*)sv;
        const unsigned long long ga = (unsigned long long)(const void*)x4;
        // D# group0: count=1 | lds_addr | global_addr[56:0] | type=2 (bits 127:126)
        u32x4 g0 = { 1u, lds_base,
                     (unsigned)(ga & 0xFFFFFFFFull),
                     (unsigned)((ga >> 32) & 0x01FFFFFFull) | (2u << 30) };
        // D# group1: data_size=2 (4B); tensor_dim0=2048 (bits 79:48);
        // tensor_dim1=1 (bits 111:80); tile_dim0=2048 (bits 127:112);
        // tile_dim1/2 unused; tensor_dim0_stride=2048 (bits 207:160)
        i32x8 g1 = { 0x00020000,
                     (int)((2048u & 0xFFFFu) << 16),
                     (int)(1u << 16),
                     (int)(2048u << 16),
                     0, 2048, 0, 0 };
        i32x4 gz = { 0, 0, 0, 0 };
#if TDM_6ARG
        i32x8 g4 = { 0, 0, 0, 0, 0, 0, 0, 0 };
        __builtin_amdgcn_tensor_load_to_lds(g0, g1, gz, gz, g4, 0);
#else
        __builtin_amdgcn_tensor_load_to_lds(g0, g1, gz, gz, 0);
#endif
    }
    for (int j = PER_T4 / 2; j < PER_T4; ++j) {      // chunks 2,3 -> bytes 8K..16K
        const int idx = tid + j * THREADS;
        __builtin_amdgcn_global_load_async_to_lds_b128(
            (AS1 v4i_vs*)(x4 + idx), (AS3 v4i_vs*)(sv4 + idx), 0, 0);
    }
    if (tid < 32) __builtin_amdgcn_s_wait_tensorcnt(0);
#if __has_builtin(__builtin_amdgcn_s_wait_asynccnt)
    __builtin_amdgcn_s_wait_asynccnt(0);
#else
    asm volatile("s_wait_asynccnt 0" ::: "memory");
#endif
    __syncthreads();   // all DMA complete & visible workgroup-wide
    for (int j = 0; j < PER_T4; ++j) {
        v4f v = sv4[tid + j * THREADS];
        loc[j] = v;
        tmax = fmaxf(tmax, fmaxf(fmaxf(v.x, v.y), fmaxf(v.z, v.w)));
    }
#elif USE_ASYNC_LDS
    // CDNA5 async DMA: global -> LDS without a VGPR round-trip (ASYNCcnt path).
    for (int j = 0; j < PER_T4; ++j) {
        const int idx = tid + j * THREADS;
        __builtin_amdgcn_global_load_async_to_lds_b128(
            (AS1 v4i_vs*)(x4 + idx), (AS3 v4i_vs*)(sv4 + idx), 0, 0);
    }
#if __has_builtin(__builtin_amdgcn_s_wait_asynccnt)
    __builtin_amdgcn_s_wait_asynccnt(0);
#else
    asm volatile("s_wait_asynccnt 0" ::: "memory");
#endif
    __syncthreads();   // every wave waited on its own asyncs; now all 16KB visible
    for (int j = 0; j < PER_T4; ++j) {
        v4f v = sv4[tid + j * THREADS];
        loc[j] = v;
        tmax = fmaxf(tmax, fmaxf(fmaxf(v.x, v.y), fmaxf(v.z, v.w)));
    }
#else
    for (int j = 0; j < PER_T4; ++j) {
        v4f v = __builtin_nontemporal_load(&x4[tid + j * THREADS]);
        loc[j] = v;
        tmax = fmaxf(tmax, fmaxf(fmaxf(v.x, v.y), fmaxf(v.z, v.w)));
    }
#endif
    // row max reduction
    for (int off = 16; off > 0; off >>= 1)
        tmax = fmaxf(tmax, __shfl_xor(tmax, off, 32));
    if ((tid & 31) == 0) redbuf[tid >> 5] = tmax;
    __syncthreads();
    if (tid == 0) {
        float m = redbuf[0];
        for (int i = 1; i < 8; ++i) m = fmaxf(m, redbuf[i]);
        s_mx = m;
    }
    __syncthreads();
    const float mx = s_mx;
    for (int j = 0; j < PER_T4; ++j) {
        v4f v = loc[j];
        v = v - mx;                       // broadcast subtract
        sv4[tid + j * THREADS] = v;
    }
    __syncthreads();

    // ---- 2) Radix-select: exact key of the 128th-largest element --------------
    unsigned prefix = 0, pmask = 0, want = KTOP;
    for (int pass = 0; pass < 4; ++pass) {
        const int shift = 24 - 8 * pass;
        hist[tid] = 0u;
        __syncthreads();
        for (int j = 0; j < PER_T; ++j) {
            unsigned k = f2key(sv[tid + j * THREADS]);
            if ((k & pmask) == prefix)
                atomicAdd(&hist[(k >> shift) & 0xFFu], 1u);
        }
        __syncthreads();
        // suffix-inclusive scan: cum[b] = sum_{j>=b} hist[j]
        cum[tid] = hist[tid];
        __syncthreads();
        for (int d = 1; d < 256; d <<= 1) {
            unsigned t = (tid + d < 256) ? cum[tid + d] : 0u;
            __syncthreads();
            cum[tid] += t;
            __syncthreads();
        }
        unsigned cnext = (tid == 255) ? 0u : cum[tid + 1];
        if (cum[tid] >= want && cnext < want) {   // unique bin containing rank `want`
            s_selb = (unsigned)tid;
            s_want = want - cnext;
        }
        __syncthreads();
        prefix |= (s_selb << shift);
        pmask  |= (0xFFu << shift);
        want    = s_want;
    }
    const unsigned Tkey = prefix;   // key of 128th largest; `want` = tie-rank to include

    // ---- 3) Gather top-128 (strict-greater + exact number of ties) ------------
    if (tid == 0) { s_gtc = 0u; s_tiec = 0u; s_scount = 0u; }
    __syncthreads();
    const unsigned base_tie = KTOP - want;
    for (int j = 0; j < PER_T; ++j) {
        float v = sv[tid + j * THREADS];
        unsigned k = f2key(v);
        if (k > Tkey) {
            unsigned p = atomicAdd(&s_gtc, 1u);
            topv[p] = v;
        } else if (k == Tkey) {
            unsigned t = atomicAdd(&s_tiec, 1u);
            if (t < want) topv[base_tie + t] = v;   // tie values are bit-identical
        }
    }
    __syncthreads();

    // ---- 4) Bitonic sort of 128 values, descending ----------------------------
    for (int ksz = 2; ksz <= KTOP; ksz <<= 1) {
        for (int j = ksz >> 1; j >= 1; j >>= 1) {
            if (tid < KTOP) {
                int ixj = tid ^ j;
                if (ixj > tid) {
                    float a = topv[tid], b = topv[ixj];
                    bool desc = (tid & ksz) == 0;
                    if (desc ? (a < b) : (a > b)) { topv[tid] = b; topv[ixj] = a; }
                }
            }
            __syncthreads();
        }
    }
    // zero-padded copy so the WMMA B fragment loads unconditionally (no saveexec)
    bpad[tid] = (tid < KTOP) ? topv[tid] : 0.0f;
    __syncthreads();

    // ---- 5) Inclusive cumsum of topv[0..127] ----------------------------------
#if USE_WMMA_SCAN
    // D(16x16) = L16(16x16 lower-tri ones) x V(16x16), V[m][j] = topv[16j+m],
    // chained as four 16x16x4 f32 WMMAs. Column j of D = within-segment prefix.
    {
        const int lane = tid & 31;
        const int m  = lane & 15;        // A: row M;  B: column j;  D: column N
        const int hi = lane >> 4;        // lane-half selects K (A/B) or M+8 (D)
        v8f acc = {};
        #pragma unroll
        for (int c = 0; c < 4; ++c) {
            const int k0 = 4 * c + 2 * hi;     // ISA 16x4 f32 A layout: K={0,1|2,3}
            v2f a;
            a.x = (k0     <= m) ? 1.0f : 0.0f; // L16[m][k0]
            a.y = (k0 + 1 <= m) ? 1.0f : 0.0f; // L16[m][k0+1]
            const v2f b = *(const v2f*)&bpad[16 * m + k0];  // V[k0..k0+1][j=m], ds_load_b64
            acc = __builtin_amdgcn_wmma_f32_16x16x4_f32(
                false, a, false, b, (short)0, acc, false, false);
        }
        // D layout: VGPR v holds row M = v + 8*hi, lane = column N
        if (tid < 32 && m < 8) {
            #pragma unroll
            for (int v = 0; v < 8; ++v)
                cs[16 * m + (v + 8 * hi)] = acc[v];
        }
    }
    __syncthreads();
    if (tid == 0) {                 // exclusive scan of the 8 segment totals
        float run = 0.0f;
        for (int j = 0; j < 8; ++j) { soff[j] = run; run += cs[16 * j + 15]; }
    }
    __syncthreads();
    if (tid < KTOP) cs[tid] += soff[tid >> 4];
    __syncthreads();
#else
    if (tid < KTOP) cs[tid] = topv[tid];
    __syncthreads();
    for (int d = 1; d < KTOP; d <<= 1) {
        float t = 0.0f;
        if (tid < KTOP && tid >= d) t = cs[tid - d];
        __syncthreads();
        if (tid < KTOP && tid >= d) cs[tid] += t;
        __syncthreads();
    }
#endif

    // ---- 6) Support size and tau ----------------------------------------------
    if (tid < KTOP) {
        float ratio = cs[tid] / (topv[tid] + EPSF);
        if ((float)(tid + 1) <= ratio) atomicAdd(&s_scount, 1u);
    }
    __syncthreads();
    if (tid == 0) {
        unsigned sup = s_scount;
        if (sup < 1u) sup = 1u;
        s_tau = cs[sup - 1] / ((float)sup + EPSF);
    }
    __syncthreads();
    const float tau = s_tau;

    // ---- 7) Output: relu(x - max - tau), streaming (non-temporal) B128 --------
    for (int j = 0; j < PER_T4; ++j) {
        v4f v = sv4[tid + j * THREADS];
        v4f o;
        o.x = fmaxf(v.x - tau, 0.0f);
        o.y = fmaxf(v.y - tau, 0.0f);
        o.z = fmaxf(v.z - tau, 0.0f);
        o.w = fmaxf(v.w - tau, 0.0f);
        __builtin_nontemporal_store(o, &o4[tid + j * THREADS]);
    }
}

extern "C" void kernel_launch(void* const* d_in, const int* in_sizes, int n_in,
                              void* d_out, int out_size, void* d_ws, size_t ws_size,
                              hipStream_t stream) {
    const float* x = (const float*)d_in[0];
    float* out = (float*)d_out;
    const int rows = in_sizes[0] / ROWLEN;   // 16*1024 = 16384
    soft_threshold_topk_kernel<<<dim3(rows), dim3(THREADS), 0, stream>>>(x, out);
}